// LinearAttention_47708496724178
// MI455X (gfx1250) — compile-verified
//
#include <hip/hip_runtime.h>
#include <hip/hip_bf16.h>

typedef __attribute__((ext_vector_type(2))) float v2f;
typedef __attribute__((ext_vector_type(8))) float v8f;

#define HEADS 4
#define DIMH  32
#define NTOK  4096          // h*w = 64*64
#define CCH   256           // input channels
#define HID   128           // HEADS*DIMH
#define QKV_O 384           // 3*HID

static __device__ __forceinline__ v8f wmma_f32x4(v2f a, v2f b, v8f c) {
    // V_WMMA_F32_16X16X4_F32 : D = A(16x4) * B(4x16) + C(16x16)
    return __builtin_amdgcn_wmma_f32_16x16x4_f32(
        /*neg_a=*/false, a, /*neg_b=*/false, b,
        /*c_mod=*/(short)0, c, /*reuse_a=*/false, /*reuse_b=*/false);
}

// ---------------------------------------------------------------------------
// Kernel 1: qkv[b,o,n] = sum_c w_qkv[o,c] * x[b,c,n]   (M=384, K=256, N=4096)
// block = 128 threads (4 waves); each wave computes a 16x64 strip (4 accums).
// ---------------------------------------------------------------------------
__global__ __launch_bounds__(128) void qkv_gemm(const float* __restrict__ x,
                                                const float* __restrict__ wqkv,
                                                float* __restrict__ qkv) {
    const int lane  = threadIdx.x & 31;
    const int wave  = threadIdx.x >> 5;
    const int b     = blockIdx.z;
    const int m0    = blockIdx.y * 16;
    const int nbase = blockIdx.x * 256 + wave * 64;
    const int l15   = lane & 15;
    const int kb    = (lane < 16) ? 0 : 2;   // K-pair base for A/B fragments
    const int rhi   = (lane < 16) ? 0 : 8;   // C/D row offset for hi lanes

    const float* xb   = x   + (size_t)b * CCH * NTOK;
    float*       outb = qkv + (size_t)b * QKV_O * NTOK;

    v8f acc[4];
#pragma unroll
    for (int t = 0; t < 4; ++t) acc[t] = (v8f){};

    const float* arow = wqkv + (size_t)(m0 + l15) * CCH + kb;
    for (int k0 = 0; k0 < CCH; k0 += 4) {
        v2f a = *(const v2f*)(arow + k0);          // A[m][k0+kb], A[m][k0+kb+1]
#pragma unroll
        for (int t = 0; t < 4; ++t) {
            const int n = nbase + t * 16 + l15;
            v2f bf;
            bf.x = xb[(size_t)(k0 + kb)     * NTOK + n];
            bf.y = xb[(size_t)(k0 + kb + 1) * NTOK + n];
            acc[t] = wmma_f32x4(a, bf, acc[t]);
        }
    }

#pragma unroll
    for (int t = 0; t < 4; ++t) {
        const int n = nbase + t * 16 + l15;
#pragma unroll
        for (int r = 0; r < 8; ++r)
            outb[(size_t)(m0 + r + rhi) * NTOK + n] = acc[t][r];
    }
}

// ---------------------------------------------------------------------------
// Kernel 2: per-row stats for softmax over the K part of qkv (rows 128..255):
//   maxs[row] = max_n k[row][n] ;  sums[row] = sum_n exp(k[row][n] - max)
// No writeback of k itself — exp is re-applied in-stream by the attn kernel
// and the 1/sum normalization is folded into the 32x32 context matrix.
// ---------------------------------------------------------------------------
__global__ __launch_bounds__(256) void k_rowstats(const float* __restrict__ qkv,
                                                  float* __restrict__ maxs,
                                                  float* __restrict__ sums) {
    const int row = blockIdx.x;           // 0 .. B*128-1
    const int b   = row >> 7;
    const int dh  = row & 127;            // h*32 + d
    const float* p = qkv + (size_t)b * QKV_O * NTOK + (size_t)(HID + dh) * NTOK;

    __shared__ float red[256];
    const int tid = threadIdx.x;

    float m = -3.402823466e38f;
    for (int i = tid; i < NTOK; i += 256) m = fmaxf(m, p[i]);
    red[tid] = m;
    __syncthreads();
    for (int s = 128; s > 0; s >>= 1) {
        if (tid < s) red[tid] = fmaxf(red[tid], red[tid + s]);
        __syncthreads();
    }
    m = red[0];
    __syncthreads();

    float sum = 0.f;
    for (int i = tid; i < NTOK; i += 256) sum += __expf(p[i] - m);
    red[tid] = sum;
    __syncthreads();
    for (int s = 128; s > 0; s >>= 1) {
        if (tid < s) red[tid] += red[tid + s];
        __syncthreads();
    }
    if (tid == 0) {
        maxs[row] = m;
        sums[row] = red[0];
    }
}

// ---------------------------------------------------------------------------
// Kernel 3 (fused): per (b,h)
//   phase1: ctx[d][e] = (1/S[d]) * sum_n exp(k[d][n]-m[d]) * v[e][n]
//   phase2: out[e][n] = sum_d ctx[d][e] * q[d][n]
// block = 256 threads (8 waves). ctx reduced + normalized via LDS.
// ---------------------------------------------------------------------------
__global__ __launch_bounds__(256) void attn_ctx_out(const float* __restrict__ qkv,
                                                    const float* __restrict__ maxs,
                                                    const float* __restrict__ sums,
                                                    float* __restrict__ outw) {
    const int bh = blockIdx.x;
    const int b  = bh >> 2;
    const int h  = bh & 3;

    const float* q  = qkv + (size_t)b * QKV_O * NTOK + (size_t)(h * DIMH) * NTOK;
    const float* ks = qkv + (size_t)b * QKV_O * NTOK + (size_t)(HID + h * DIMH) * NTOK;
    const float* v  = qkv + (size_t)b * QKV_O * NTOK + (size_t)(2 * HID + h * DIMH) * NTOK;
    const float* mrow = maxs + (size_t)b * HID + h * DIMH;   // m[d], d=0..31
    const float* srow = sums + (size_t)b * HID + h * DIMH;   // S[d], d=0..31
    float* outp     = outw + (size_t)b * HID * NTOK + (size_t)(h * DIMH) * NTOK;

    __shared__ float part[8][1024];   // per-wave 32x32 partial context
    __shared__ float ctx[1024];       // reduced context, ctx[d*32+e]

    const int lane = threadIdx.x & 31;
    const int wave = threadIdx.x >> 5;
    const int l15  = lane & 15;
    const int kb   = (lane < 16) ? 0 : 2;
    const int rhi  = (lane < 16) ? 0 : 8;

    // ---- phase 1: each wave covers K-chunk of 512 along n ----
    const float am0 = mrow[l15];        // row max for d = l15
    const float am1 = mrow[16 + l15];   // row max for d = 16 + l15

    v8f acc[4];
#pragma unroll
    for (int t = 0; t < 4; ++t) acc[t] = (v8f){};

    const int kstart = wave * (NTOK / 8);
    for (int k0 = kstart; k0 < kstart + NTOK / 8; k0 += 4) {
        // A[m=d][k=n] = exp(ks[d*N+n] - m[d]) ; B[k=n][col=e] = v[e*N + n]
        v2f a0 = *(const v2f*)(ks + (size_t)(0  + l15) * NTOK + k0 + kb);
        v2f a1 = *(const v2f*)(ks + (size_t)(16 + l15) * NTOK + k0 + kb);
        v2f b0 = *(const v2f*)(v  + (size_t)(0  + l15) * NTOK + k0 + kb);
        v2f b1 = *(const v2f*)(v  + (size_t)(16 + l15) * NTOK + k0 + kb);
        a0.x = __expf(a0.x - am0);  a0.y = __expf(a0.y - am0);
        a1.x = __expf(a1.x - am1);  a1.y = __expf(a1.y - am1);
        acc[0] = wmma_f32x4(a0, b0, acc[0]);   // (mt=0, nt=0)
        acc[1] = wmma_f32x4(a0, b1, acc[1]);   // (mt=0, nt=1)
        acc[2] = wmma_f32x4(a1, b0, acc[2]);   // (mt=1, nt=0)
        acc[3] = wmma_f32x4(a1, b1, acc[3]);   // (mt=1, nt=1)
    }

#pragma unroll
    for (int t = 0; t < 4; ++t) {
        const int mt = t >> 1, nt = t & 1;
#pragma unroll
        for (int r = 0; r < 8; ++r) {
            const int row = mt * 16 + r + rhi;   // d
            const int col = nt * 16 + l15;       // e
            part[wave][row * 32 + col] = acc[t][r];
        }
    }
    __syncthreads();

    // reduce partials and fold in the softmax 1/sum normalization per d-row
    for (int i = threadIdx.x; i < 1024; i += 256) {
        float s = 0.f;
#pragma unroll
        for (int w = 0; w < 8; ++w) s += part[w][i];
        ctx[i] = s * (1.0f / srow[i >> 5]);
    }
    __syncthreads();

    // ---- phase 2: out = ctx^T @ q ----
    // hoist A-fragments from LDS: A[e][d] = ctx[d*32 + e]
    v2f afrag[2][8];
#pragma unroll
    for (int mt = 0; mt < 2; ++mt) {
#pragma unroll
        for (int kk = 0; kk < 8; ++kk) {
            const int e  = mt * 16 + l15;
            const int kd = kk * 4 + kb;
            afrag[mt][kk].x = ctx[(kd)     * 32 + e];
            afrag[mt][kk].y = ctx[(kd + 1) * 32 + e];
        }
    }

    // 512 output tiles (2 x 256), distributed round-robin over 8 waves
    for (int t = wave; t < 512; t += 8) {
        const int mt = t & 1;
        const int n0 = (t >> 1) * 16;
        v8f o = (v8f){};
#pragma unroll
        for (int kk = 0; kk < 8; ++kk) {
            const int kd = kk * 4 + kb;
            v2f bf;
            const int n = n0 + l15;
            bf.x = q[(size_t)(kd)     * NTOK + n];
            bf.y = q[(size_t)(kd + 1) * NTOK + n];
            o = wmma_f32x4(afrag[mt][kk], bf, o);
        }
        const int m0 = mt * 16;
#pragma unroll
        for (int r = 0; r < 8; ++r)
            outp[(size_t)(m0 + r + rhi) * NTOK + n0 + l15] = o[r];
    }
}

// ---------------------------------------------------------------------------
// Kernel 4: y[b,o,n] = sum_c w_out[o,c] * out[b,c,n] + b_out[o]
//           (M=256, K=128, N=4096) — same tiling as kernel 1, bias fused.
// ---------------------------------------------------------------------------
__global__ __launch_bounds__(128) void proj_out(const float* __restrict__ outw,
                                                const float* __restrict__ wout,
                                                const float* __restrict__ bout,
                                                float* __restrict__ y) {
    const int lane  = threadIdx.x & 31;
    const int wave  = threadIdx.x >> 5;
    const int b     = blockIdx.z;
    const int m0    = blockIdx.y * 16;
    const int nbase = blockIdx.x * 256 + wave * 64;
    const int l15   = lane & 15;
    const int kb    = (lane < 16) ? 0 : 2;
    const int rhi   = (lane < 16) ? 0 : 8;

    const float* ob = outw + (size_t)b * HID * NTOK;
    float*       yb = y    + (size_t)b * CCH * NTOK;

    v8f acc[4];
#pragma unroll
    for (int t = 0; t < 4; ++t) acc[t] = (v8f){};

    const float* arow = wout + (size_t)(m0 + l15) * HID + kb;
    for (int k0 = 0; k0 < HID; k0 += 4) {
        v2f a = *(const v2f*)(arow + k0);
#pragma unroll
        for (int t = 0; t < 4; ++t) {
            const int n = nbase + t * 16 + l15;
            v2f bf;
            bf.x = ob[(size_t)(k0 + kb)     * NTOK + n];
            bf.y = ob[(size_t)(k0 + kb + 1) * NTOK + n];
            acc[t] = wmma_f32x4(a, bf, acc[t]);
        }
    }

#pragma unroll
    for (int t = 0; t < 4; ++t) {
        const int n = nbase + t * 16 + l15;
#pragma unroll
        for (int r = 0; r < 8; ++r) {
            const int row = m0 + r + rhi;
            yb[(size_t)row * NTOK + n] = acc[t][r] + bout[row];
        }
    }
}

// ---------------------------------------------------------------------------
extern "C" void kernel_launch(void* const* d_in, const int* in_sizes, int n_in,
                              void* d_out, int out_size, void* d_ws, size_t ws_size,
                              hipStream_t stream) {
    (void)n_in; (void)out_size; (void)ws_size;

    const float* x    = (const float*)d_in[0];   // (B, 256, 64, 64)
    const float* wqkv = (const float*)d_in[1];   // (384, 256)
    const float* wout = (const float*)d_in[2];   // (256, 128)
    const float* bout = (const float*)d_in[3];   // (256,)
    float*       y    = (float*)d_out;           // (B, 256, 64, 64)

    const int B = in_sizes[0] / (CCH * NTOK);    // 16

    float* qkv  = (float*)d_ws;                             // B*384*4096 floats
    float* outw = qkv  + (size_t)B * QKV_O * NTOK;          // B*128*4096 floats
    float* maxs = outw + (size_t)B * HID   * NTOK;          // B*128 floats
    float* sums = maxs + (size_t)B * HID;                   // B*128 floats

    qkv_gemm    <<<dim3(NTOK / 256, QKV_O / 16, B), 128, 0, stream>>>(x, wqkv, qkv);
    k_rowstats  <<<dim3(B * HID),                   256, 0, stream>>>(qkv, maxs, sums);
    attn_ctx_out<<<dim3(B * HEADS),                 256, 0, stream>>>(qkv, maxs, sums, outw);
    proj_out    <<<dim3(NTOK / 256, CCH / 16, B),   128, 0, stream>>>(outw, wout, bout, y);
}